// SMFNet_74406013436260
// MI455X (gfx1250) — compile-verified
//
#include <hip/hip_runtime.h>
#include <hip/hip_bf16.h>

// Problem constants (match reference)
#define BB 2
#define NN 8192
#define DD 64
#define DV 64
#define MM 3

typedef __attribute__((ext_vector_type(2))) float v2f;
typedef __attribute__((ext_vector_type(8))) float v8f;

// ---------------------------------------------------------------------------
// Kernel 1: V0 = X @ Wg + bg   via V_WMMA_F32_16X16X4_F32 (full fp32).
// One wave (32 threads) per 16-row tile. K = 64 = 16 chained k-steps of 4.
// A layout (16x4 f32): lane = M (r), vgpr0 = K {0,2}, vgpr1 = K {1,3}; lane
// half (hi) selects K-offset +2. B layout mirrored: lane = N col, vgprs = K.
// D layout: vgpr j -> row (j + 8*hi), lane -> col r.
// ---------------------------------------------------------------------------
__global__ __launch_bounds__(32) void v0_gemm_kernel(
    const float* __restrict__ X, const float* __restrict__ Wg,
    const float* __restrict__ bg, float* __restrict__ V0) {
  const int tile = blockIdx.x;               // 0 .. B*N/16 - 1
  const int b    = tile / (NN / 16);
  const int row0 = (tile % (NN / 16)) * 16;
  const int lane = threadIdx.x;
  const int r    = lane & 15;
  const int hi   = lane >> 4;

  // Preload this wave's A tile: 16 k-steps, v2f each (32 VGPRs total).
  const float* xrow = X + ((size_t)b * NN + row0 + r) * DD;
  v2f areg[16];
#pragma unroll
  for (int kk = 0; kk < 16; ++kk) {
    const int k0 = 4 * kk + 2 * hi;
    areg[kk] = *(const v2f*)(xrow + k0);     // 8B-aligned (k0 even)
  }

#pragma unroll
  for (int ct = 0; ct < 4; ++ct) {           // DV = 64 -> 4 column tiles
    const int col = 16 * ct + r;
    v8f acc = {0.f, 0.f, 0.f, 0.f, 0.f, 0.f, 0.f, 0.f};
#pragma unroll
    for (int kk = 0; kk < 16; ++kk) {
      const int k0 = 4 * kk + 2 * hi;
      v2f bm;
      bm.x = Wg[(size_t)k0 * DV + col];
      bm.y = Wg[(size_t)(k0 + 1) * DV + col];
      acc = __builtin_amdgcn_wmma_f32_16x16x4_f32(
          /*neg_a=*/false, areg[kk], /*neg_b=*/false, bm,
          /*c_mod=*/(short)0, acc, /*reuse_a=*/false, /*reuse_b=*/false);
    }
    const float bias = bg[col];
#pragma unroll
    for (int j = 0; j < 8; ++j) {
      V0[((size_t)b * NN + row0 + j + 8 * hi) * DV + col] = acc[j] + bias;
    }
  }
}

// ---------------------------------------------------------------------------
// Kernel 2: bidiagonal coefficients of the chord-masked W matrices.
//   a[b,m,i] = X[b,i,:] . Wf[m][:,i]         + bf[m][i]
//   c[b,m,i] = X[b,i,:] . Wf[m][:,(i+1)%N]   + bf[m][(i+1)%N]
// X tile staged through LDS (stride 65 -> no bank conflicts); Wf column
// accesses are coalesced (column index == thread index).
// ---------------------------------------------------------------------------
__global__ __launch_bounds__(128) void coeff_kernel(
    const float* __restrict__ X, const float* __restrict__ Wf,
    const float* __restrict__ bf, float* __restrict__ ca,
    float* __restrict__ cc) {
  __shared__ float Xs[128 * 65];
  const int tid  = threadIdx.x;
  const int cpb  = NN / 128;                 // chunks per (b,m) image
  const int bi   = blockIdx.x;
  const int i0   = (bi % cpb) * 128;
  const int m    = (bi / cpb) % MM;
  const int b    = bi / (cpb * MM);

  // Cooperative, coalesced load of 128 X rows into LDS.
  const float* xbase = X + ((size_t)b * NN + i0) * DD;
  for (int idx = tid; idx < 128 * DD; idx += 128) {
    const int row = idx >> 6;
    const int col = idx & 63;
    Xs[row * 65 + col] = xbase[idx];
  }
  __syncthreads();

  const int i  = i0 + tid;
  const int i1 = (i + 1) & (NN - 1);
  float accA = bf[(size_t)m * NN + i];
  float accC = bf[(size_t)m * NN + i1];
  const float* wcol = Wf + (size_t)m * DD * NN;
#pragma unroll 8
  for (int d = 0; d < DD; ++d) {
    const float x = Xs[tid * 65 + d];
    accA += x * wcol[(size_t)d * NN + i];
    accC += x * wcol[(size_t)d * NN + i1];
  }
  const size_t o = ((size_t)b * MM + m) * NN + i;
  ca[o] = accA;
  cc[o] = accC;
}

// ---------------------------------------------------------------------------
// Kernel 3: apply the M=3 bidiagonal chain.
//   V1[j] = a0[j] V0[j] + c0[j] V0[j+1]
//   V2[j] = a1[j] V1[j] + c1[j] V1[j+1]
//   out[i] = a2[i] V2[i] + c2[i] V2[i+1]
// One 64-thread block per (b, i); thread = dv lane. V0 reads coalesced.
// ---------------------------------------------------------------------------
__global__ __launch_bounds__(64) void combine_kernel(
    const float* __restrict__ V0, const float* __restrict__ ca,
    const float* __restrict__ cc, float* __restrict__ out) {
  const int dv = threadIdx.x;
  const int b  = blockIdx.x >> 13;           // / N
  const int i  = blockIdx.x & (NN - 1);

  const int j0 = i;
  const int j1 = (i + 1) & (NN - 1);
  const int j2 = (i + 2) & (NN - 1);
  const int j3 = (i + 3) & (NN - 1);

  const size_t cb = (size_t)b * MM * NN;
  const float a0_0 = ca[cb + 0 * NN + j0], c0_0 = cc[cb + 0 * NN + j0];
  const float a0_1 = ca[cb + 0 * NN + j1], c0_1 = cc[cb + 0 * NN + j1];
  const float a0_2 = ca[cb + 0 * NN + j2], c0_2 = cc[cb + 0 * NN + j2];
  const float a1_0 = ca[cb + 1 * NN + j0], c1_0 = cc[cb + 1 * NN + j0];
  const float a1_1 = ca[cb + 1 * NN + j1], c1_1 = cc[cb + 1 * NN + j1];
  const float a2_0 = ca[cb + 2 * NN + j0], c2_0 = cc[cb + 2 * NN + j0];

  const size_t vb = (size_t)b * NN;
  const float v0 = V0[(vb + j0) * DV + dv];
  const float v1 = V0[(vb + j1) * DV + dv];
  const float v2 = V0[(vb + j2) * DV + dv];
  const float v3 = V0[(vb + j3) * DV + dv];

  const float u0 = a0_0 * v0 + c0_0 * v1;
  const float u1 = a0_1 * v1 + c0_1 * v2;
  const float u2 = a0_2 * v2 + c0_2 * v3;
  const float w0 = a1_0 * u0 + c1_0 * u1;
  const float w1 = a1_1 * u1 + c1_1 * u2;

  out[(vb + i) * DV + dv] = a2_0 * w0 + c2_0 * w1;
}

extern "C" void kernel_launch(void* const* d_in, const int* in_sizes, int n_in,
                              void* d_out, int out_size, void* d_ws, size_t ws_size,
                              hipStream_t stream) {
  const float* X  = (const float*)d_in[0];   // (B, N, D)
  const float* Wg = (const float*)d_in[1];   // (D, DV)
  const float* bg = (const float*)d_in[2];   // (DV,)
  const float* Wf = (const float*)d_in[3];   // (M, D, N)
  const float* bf = (const float*)d_in[4];   // (M, N)
  float* out = (float*)d_out;                // (B, N, DV)

  // Workspace: V0 (B*N*DV) | a (B*M*N) | c (B*M*N)   ~4.6 MB total
  float* V0 = (float*)d_ws;
  float* ca = V0 + (size_t)BB * NN * DV;
  float* cc = ca + (size_t)BB * MM * NN;

  v0_gemm_kernel<<<BB * NN / 16, 32, 0, stream>>>(X, Wg, bg, V0);
  coeff_kernel<<<BB * MM * NN / 128, 128, 0, stream>>>(X, Wf, bf, ca, cc);
  combine_kernel<<<BB * NN, 64, 0, stream>>>(V0, ca, cc, out);
}